// Capsule_34677565948479
// MI455X (gfx1250) — compile-verified
//
#include <hip/hip_runtime.h>

#define NUMC 10
#define D0   60
#define D1   16
#define BATCH 32
#define HH 32
#define WW 32
#define CH 128
#define P     (HH * WW)          // 1024 (GEMM K)
#define NOUT  (NUMC * D0 * D1)   // 9600 (GEMM N)
#define MROWS (BATCH * CH)       // 4096 (GEMM M)
#define DD    (D0 * D1)          // 960
#define KC    32

typedef __attribute__((ext_vector_type(2))) float v2f;
typedef __attribute__((ext_vector_type(8))) float v8f;

#define WMMA_F32(a, b, c) \
    __builtin_amdgcn_wmma_f32_16x16x4_f32(false, (a), false, (b), (short)0, (c), false, false)

// ---------------------------------------------------------------------------
// Kernel 1: 2x2 SAME conv as an implicit-im2col WMMA GEMM:
//   y[32768,128] = A[32768,512] x W[512,128],  A[s, tap*128+ci] =
//   x[s + kh*32 + kw, ci] masked by (h+kh<32 && w+kw<32) (TF SAME, pad after).
// Block = 8 waves: 4 M-tiles (16 rows) x 2 N-slabs (64 co). W chunks are
// pair-staged in LDS (stride 288 dwords == 32 mod 64 banks -> conflict-free
// ds_load_b64). Invalid taps: clamp address to a safe row, cndmask pair to 0.
// ---------------------------------------------------------------------------
__global__ __launch_bounds__(256) void conv_wmma(const float* __restrict__ x,
                                                 const float* __restrict__ Wc,
                                                 float* __restrict__ y) {
    __shared__ float bs[(KC / 2) * 288];

    const int wave = threadIdx.x >> 5;
    const int lane = threadIdx.x & 31;
    const int m16  = lane & 15;
    const int klo  = (lane >> 4) << 1;                 // 0 or 2
    const int mw   = wave >> 1;                        // 0..3
    const int col0 = (wave & 1) << 6;                  // 0 or 64
    const int s    = blockIdx.x * 64 + mw * 16 + m16;  // spatial row (b*1024+h*32+w)
    const int hq   = (s >> 5) & 31;
    const int wq   = s & 31;

    v8f acc0 = {}, acc1 = {}, acc2 = {}, acc3 = {};

    #pragma unroll
    for (int tap = 0; tap < 4; ++tap) {
        const int  kh = tap >> 1, kw = tap & 1;
        const bool v  = (hq + kh < HH) && (wq + kw < WW);
        const float* ap = x + (size_t)(v ? (s + kh * WW + kw) : s) * CH + klo;

        #pragma unroll
        for (int kc = 0; kc < CH; kc += KC) {
            const int kb = tap * CH + kc;              // global K base
            __syncthreads();
            for (int i = threadIdx.x; i < (KC / 2) * CH; i += 256) {
                int kp = i >> 7, nn = i & (CH - 1);
                v2f t = { Wc[(size_t)(kb + 2 * kp)     * CH + nn],
                          Wc[(size_t)(kb + 2 * kp + 1) * CH + nn] };
                *(v2f*)&bs[kp * 288 + 2 * nn] = t;
            }
            __syncthreads();

            #pragma unroll
            for (int kk = 0; kk < KC; kk += 4) {
                v2f ar = *(const v2f*)(ap + kc + kk);
                v2f a  = v ? ar : (v2f){0.f, 0.f};
                const float* bp = &bs[((kk + klo) >> 1) * 288 + 2 * (col0 + m16)];
                v2f b0 = *(const v2f*)(bp);
                v2f b1 = *(const v2f*)(bp + 32);
                v2f b2 = *(const v2f*)(bp + 64);
                v2f b3 = *(const v2f*)(bp + 96);
                acc0 = WMMA_F32(a, b0, acc0);
                acc1 = WMMA_F32(a, b1, acc1);
                acc2 = WMMA_F32(a, b2, acc2);
                acc3 = WMMA_F32(a, b3, acc3);
            }
        }
    }

    const int rbase = blockIdx.x * 64 + mw * 16 + ((lane >> 4) << 3);
    float* cp = y + (size_t)rbase * CH + col0 + m16;
    #pragma unroll
    for (int j = 0; j < 8; ++j) {
        cp[0]  = acc0[j];
        cp[16] = acc1[j];
        cp[32] = acc2[j];
        cp[48] = acc3[j];
        cp += CH;
    }
}

// ---------------------------------------------------------------------------
// Kernel 2: fp32 GEMM via V_WMMA_F32_16X16X4_F32 with double-buffered
// GLOBAL_LOAD_ASYNC_TO_LDS staging (ASYNCcnt + barrier protocol):
//   C[4096,9600] = A[4096,1024] * B[1024,9600]
// Block = 8 waves stacked in M (128 rows); wave owns 16x64. B pair-staged in
// LDS (stride 160 dwords == 32 mod 64 banks -> conflict-free ds_load_b64).
// Per chunk: issue async DMA for chunk k+1 into buf^1, WMMA-compute chunk k
// from buf, then s_wait_asynccnt 0 + barrier (RAW for next chunk, WAR for the
// buffer being refilled). DMA runs fully under WMMA compute, no VGPR bounce.
// ---------------------------------------------------------------------------
__global__ __launch_bounds__(256) void gemm_wmma_f32(const float* __restrict__ A,
                                                     const float* __restrict__ Bm,
                                                     float* __restrict__ C) {
    __shared__ float bs[2][(KC / 2) * 160];

    const int wave = threadIdx.x >> 5;
    const int lane = threadIdx.x & 31;
    const int m16  = lane & 15;
    const int klo  = (lane >> 4) << 1;
    const int row  = blockIdx.y * 128 + wave * 16 + m16;
    const int col0 = blockIdx.x * 64;

    v8f acc0 = {}, acc1 = {}, acc2 = {}, acc3 = {};
    const float* aPtr = A + (size_t)row * P + klo;

    // async-DMA stage of B[kb:kb+32, col0:col0+64] into bs[bf], K-pair layout
    auto stage = [&](int bf, int kb) {
        #pragma unroll
        for (int t = 0; t < 4; ++t) {                  // 4*256 = 1024 pairs
            int i  = (int)threadIdx.x + t * 256;
            int kp = i >> 6, nn = i & 63;
            unsigned lds0 = (unsigned)(uintptr_t)&bs[bf][kp * 160 + 2 * nn];
            unsigned lds1 = lds0 + 4u;
            unsigned off0 = (unsigned)(((kb + 2 * kp) * NOUT + col0 + nn) * 4);
            unsigned off1 = off0 + (unsigned)(NOUT * 4);
            asm volatile("global_load_async_to_lds_b32 %0, %1, %2"
                         :: "v"(lds0), "v"(off0), "s"(Bm) : "memory");
            asm volatile("global_load_async_to_lds_b32 %0, %1, %2"
                         :: "v"(lds1), "v"(off1), "s"(Bm) : "memory");
        }
    };

    stage(0, 0);
    asm volatile("s_wait_asynccnt 0x0" ::: "memory");
    __syncthreads();

    const int NCHUNK = P / KC;                         // 32
    for (int c = 0; c < NCHUNK; ++c) {
        const int bf = c & 1;
        const int kb = c * KC;
        if (c + 1 < NCHUNK) stage(bf ^ 1, kb + KC);    // DMA next chunk under compute

        #pragma unroll
        for (int kk = 0; kk < KC; kk += 4) {
            v2f a = *(const v2f*)(aPtr + kb + kk);     // {A[row,k+klo], A[row,k+klo+1]}
            const float* bp = &bs[bf][((kk + klo) >> 1) * 160 + 2 * m16];
            v2f b0 = *(const v2f*)(bp);
            v2f b1 = *(const v2f*)(bp + 32);
            v2f b2 = *(const v2f*)(bp + 64);
            v2f b3 = *(const v2f*)(bp + 96);
            acc0 = WMMA_F32(a, b0, acc0);
            acc1 = WMMA_F32(a, b1, acc1);
            acc2 = WMMA_F32(a, b2, acc2);
            acc3 = WMMA_F32(a, b3, acc3);
        }

        asm volatile("s_wait_asynccnt 0x0" ::: "memory");
        __syncthreads();
    }

    const int rbase = blockIdx.y * 128 + wave * 16 + ((lane >> 4) << 3);
    float* cp = C + (size_t)rbase * NOUT + col0 + m16;
    #pragma unroll
    for (int j = 0; j < 8; ++j) {
        cp[0]  = acc0[j];
        cp[16] = acc1[j];
        cp[32] = acc2[j];
        cp[48] = acc3[j];
        cp += NOUT;
    }
}

// ---------------------------------------------------------------------------
// Kernel 3: dynamic routing, one workgroup per batch element.
// u_hat (157MB) is L2-resident on MI455X (192MB L2): repeated passes are
// cheap. LDS: b[10][128], c[10][128], o[9600], scales (~49KB of 320KB WGP).
// ---------------------------------------------------------------------------
__global__ __launch_bounds__(256) void routing_kernel(const float* __restrict__ uhat,
                                                      float* __restrict__ out) {
    __shared__ float b_l[NUMC][CH];
    __shared__ float c_l[NUMC][CH];
    __shared__ float o_l[NUMC * DD];
    __shared__ float scale_l[NUMC];

    const int tid = threadIdx.x;
    const int bq  = blockIdx.x;
    const float* uh = uhat + (size_t)bq * CH * NOUT;   // uh[n*NOUT + (i*960+jk)]

    for (int i = tid; i < NUMC * CH; i += 256) (&b_l[0][0])[i] = 0.f;
    __syncthreads();

    for (int it = 0; it < 3; ++it) {
        // softmax over the capsule axis (i) for each n
        if (tid < CH) {
            float mx = b_l[0][tid];
            #pragma unroll
            for (int i = 1; i < NUMC; ++i) mx = fmaxf(mx, b_l[i][tid]);
            float e[NUMC];
            float s = 0.f;
            #pragma unroll
            for (int i = 0; i < NUMC; ++i) { e[i] = __expf(b_l[i][tid] - mx); s += e[i]; }
            float inv = 1.f / s;
            #pragma unroll
            for (int i = 0; i < NUMC; ++i) c_l[i][tid] = e[i] * inv;
        }
        __syncthreads();

        // o[i,jk] = sum_n c[i][n] * uhat[n, i*960+jk]  (coalesced across lanes)
        for (int idx = tid; idx < NUMC * DD; idx += 256) {
            int i = idx / DD;
            const float* up = uh + idx;
            float acc = 0.f;
            #pragma unroll 8
            for (int n = 0; n < CH; ++n)
                acc = fmaf(c_l[i][n], up[(size_t)n * NOUT], acc);
            o_l[idx] = acc;
        }
        __syncthreads();

        if (it < 2) {
            // per-capsule L2 norm of o
            if (tid < NUMC) scale_l[tid] = 0.f;
            __syncthreads();
            for (int idx = tid; idx < NUMC * DD; idx += 256) {
                float v = o_l[idx];
                atomicAdd(&scale_l[idx / DD], v * v);   // ds_add_f32
            }
            __syncthreads();
            if (tid < NUMC) scale_l[tid] = rsqrtf(fmaxf(scale_l[tid], 1e-12f));
            __syncthreads();

            // b[i][n] = scale[i] * sum_jk o[i,jk] * uhat[n, i*960+jk]
            const int wv = tid >> 5, lane = tid & 31;
            for (int p = wv; p < NUMC * CH; p += 8) {
                int i = p / CH, n = p % CH;
                const float* up = uh + (size_t)n * NOUT + i * DD;
                const float* op = &o_l[i * DD];
                float s = 0.f;
                for (int jk = lane; jk < DD; jk += 32)
                    s = fmaf(op[jk], up[jk], s);
                #pragma unroll
                for (int off = 16; off > 0; off >>= 1)
                    s += __shfl_down(s, off, 32);
                if (lane == 0) b_l[i][n] = s * scale_l[i];
            }
            __syncthreads();
        } else {
            for (int idx = tid; idx < NUMC * DD; idx += 256)
                out[(size_t)bq * (NUMC * DD) + idx] = o_l[idx];
        }
    }
}

// ---------------------------------------------------------------------------
extern "C" void kernel_launch(void* const* d_in, const int* in_sizes, int n_in,
                              void* d_out, int out_size, void* d_ws, size_t ws_size,
                              hipStream_t stream) {
    (void)in_sizes; (void)n_in; (void)out_size; (void)ws_size;

    const float* u_vecs = (const float*)d_in[0];   // (32,32,32,128)
    const float* W_conv = (const float*)d_in[1];   // (2,2,128,128) == (512,128) row-major
    const float* kern   = (const float*)d_in[2];   // (1024,9600)
    float* out = (float*)d_out;                    // (32,10,60,16)

    // workspace: conv output u (== GEMM A via raw reshape), then u_hat
    float* u    = (float*)d_ws;                               // 4,194,304 floats
    float* uhat = (float*)d_ws + (size_t)MROWS * P;           // 39,321,600 floats

    conv_wmma<<<dim3((BATCH * HH * WW) / 64), 256, 0, stream>>>(u_vecs, W_conv, u);
    gemm_wmma_f32<<<dim3(NOUT / 64, MROWS / 128), 256, 0, stream>>>(u, kern, uhat);
    routing_kernel<<<dim3(BATCH), 256, 0, stream>>>(uhat, out);
}